// ECODQN_layer_67405216744187
// MI455X (gfx1250) — compile-verified
//
#include <hip/hip_runtime.h>

#define ND 64
#define ASTRIDE 132   // 16x128 LDS tile padded: 132 % 64 = 4 -> conflict-free column reads

typedef __attribute__((ext_vector_type(2))) float v2f;
typedef __attribute__((ext_vector_type(8))) float v8f;

// Guaranteed-native CDNA5 FP32 global atomic add (no-return form, STOREcnt-tracked).
// Avoids any chance of a CAS-loop expansion on the hot scatter path.
__device__ __forceinline__ void global_fadd(float* addr, float val) {
    asm volatile("global_atomic_add_f32 %0, %1, off"
                 :
                 : "v"(addr), "v"(val)
                 : "memory");
}

// ---------------- kernel 1: zero the scatter workspace ----------------
__global__ void zero_ws_kernel(float* __restrict__ ws, int n) {
    int i = blockIdx.x * blockDim.x + threadIdx.x;
    if (i < n) ws[i] = 0.0f;
}

// ---------------- kernel 2: edge gather + atomic scatter-sum ----------------
// one wave32 per edge; each lane handles 2 consecutive features (float2)
__global__ void edge_scatter_kernel(const float* __restrict__ x,
                                    const long long* __restrict__ ei,   // [2,E] int64
                                    const float* __restrict__ ea,       // [E]
                                    float* __restrict__ summed,         // [N,64]
                                    float* __restrict__ counts,         // [N]
                                    int E) {
    int gid  = blockIdx.x * blockDim.x + threadIdx.x;
    int e    = gid >> 5;
    int lane = gid & 31;
    if (e >= E) return;
    long long c = ei[e];          // col = source node
    long long r = ei[E + e];      // row = destination node
    float w = ea[e];
    float2 xv = *((const float2*)(x + c * ND) + lane);
    float* dst = summed + r * ND + lane * 2;
    global_fadd(dst,     w * xv.x);
    global_fadd(dst + 1, w * xv.y);
    if (lane == 0)
        global_fadd(counts + r, 1.0f);
}

// ---------------- kernel 3: fused mean + two MLPs via f32 WMMA ----------------
// block = 128 threads (4 waves) handles a tile of 16 nodes.
// Wave w computes output columns [16w, 16w+16) of both 16x128 @ 128x64 GEMMs.
__global__ void node_mlp_kernel(const float* __restrict__ x,
                                const float* __restrict__ emb,
                                const float* __restrict__ summed,
                                const float* __restrict__ counts,
                                const float* __restrict__ Wmsg,  // [128,64] row-major [in][out]
                                const float* __restrict__ bmsg,  // [64]
                                const float* __restrict__ Wupd,  // [128,64]
                                const float* __restrict__ bupd,  // [64]
                                float* __restrict__ out,         // [N,64]
                                int N) {
    __shared__ float A[16 * ASTRIDE];  // 16 x 128 concat operand, padded

    const int n0   = blockIdx.x * 16;
    const int tid  = threadIdx.x;
    const int wave = tid >> 5;
    const int lane = tid & 31;
    const int l16  = lane & 15;
    const int hi   = lane >> 4;            // lane half: selects K offset 0 or 2
    const int col  = wave * 16 + l16;      // output column handled by this lane

    // ---- stage A-buffer with [x_agg | x_agg_emb] (16 x 128) ----
    for (int i = 0; i < 16; ++i) {
        int e = tid * 16 + i;              // 128 threads * 16 = 2048 = 16*128
        int r = e >> 7;
        int c = e & 127;
        float v;
        long long node = (long long)(n0 + r);
        if (c < 64) {
            float cnt = counts[node];
            v = summed[node * ND + c] / fmaxf(cnt, 1.0f);
        } else {
            v = emb[node * ND + (c - 64)];
        }
        A[r * ASTRIDE + c] = v;
    }
    __syncthreads();

    // ---- GEMM 1: [x_agg|emb] @ W_msg, K=128 in 32 steps of V_WMMA_F32_16X16X4_F32 ----
    v8f acc = {};
#pragma unroll
    for (int k = 0; k < 128; k += 4) {
        int ka = k + 2 * hi;
        v2f a; a.x = A[l16 * ASTRIDE + ka];      a.y = A[l16 * ASTRIDE + ka + 1];
        v2f b; b.x = Wmsg[ka * ND + col];        b.y = Wmsg[(ka + 1) * ND + col];
        acc = __builtin_amdgcn_wmma_f32_16x16x4_f32(false, a, false, b, (short)0, acc,
                                                    false, false);
    }
    float bias = bmsg[col];
    float m[8];
#pragma unroll
    for (int r = 0; r < 8; ++r) m[r] = fmaxf(acc[r] + bias, 0.0f);
    __syncthreads();   // everyone done reading A before overwrite

    // ---- restage A-buffer with [x | m] ----
    for (int i = 0; i < 8; ++i) {
        int e = tid * 8 + i;               // 128 threads * 8 = 1024 = 16*64
        int r = e >> 6;
        int c = e & 63;
        A[r * ASTRIDE + c] = x[(long long)(n0 + r) * ND + c];
    }
#pragma unroll
    for (int r = 0; r < 8; ++r)            // C layout: VGPR r -> rows r (lo half) / r+8 (hi half)
        A[(r + 8 * hi) * ASTRIDE + 64 + col] = m[r];
    __syncthreads();

    // ---- GEMM 2: [x|m] @ W_upd ----
    v8f acc2 = {};
#pragma unroll
    for (int k = 0; k < 128; k += 4) {
        int ka = k + 2 * hi;
        v2f a; a.x = A[l16 * ASTRIDE + ka];      a.y = A[l16 * ASTRIDE + ka + 1];
        v2f b; b.x = Wupd[ka * ND + col];        b.y = Wupd[(ka + 1) * ND + col];
        acc2 = __builtin_amdgcn_wmma_f32_16x16x4_f32(false, a, false, b, (short)0, acc2,
                                                     false, false);
    }
    float bias2 = bupd[col];
#pragma unroll
    for (int r = 0; r < 8; ++r) {
        long long node = (long long)(n0 + r + 8 * hi);
        out[node * ND + col] = fmaxf(acc2[r] + bias2, 0.0f);
    }
}

// ---------------- launcher ----------------
extern "C" void kernel_launch(void* const* d_in, const int* in_sizes, int n_in,
                              void* d_out, int out_size, void* d_ws, size_t ws_size,
                              hipStream_t stream) {
    const float*     x    = (const float*)d_in[0];
    const long long* ei   = (const long long*)d_in[1];
    const float*     ea   = (const float*)d_in[2];
    const float*     emb  = (const float*)d_in[3];
    const float*     Wmsg = (const float*)d_in[4];
    const float*     bmsg = (const float*)d_in[5];
    const float*     Wupd = (const float*)d_in[6];
    const float*     bupd = (const float*)d_in[7];
    float*           out  = (float*)d_out;

    const int N = in_sizes[0] / ND;   // 50000
    const int E = in_sizes[2];        // 800000

    float* summed = (float*)d_ws;                 // [N,64]
    float* counts = summed + (size_t)N * ND;      // [N]

    int zn = N * ND + N;
    zero_ws_kernel<<<(zn + 255) / 256, 256, 0, stream>>>(summed, zn);

    int scatter_threads = E * 32;                 // one wave32 per edge
    edge_scatter_kernel<<<(scatter_threads + 255) / 256, 256, 0, stream>>>(
        x, ei, ea, summed, counts, E);

    node_mlp_kernel<<<N / 16, 128, 0, stream>>>(
        x, emb, summed, counts, Wmsg, bmsg, Wupd, bupd, out, N);
}